// GRU_29961691857513
// MI455X (gfx1250) — compile-verified
//
#include <hip/hip_runtime.h>
#include <hip/hip_bf16.h>

// Vanilla tanh RNN on gfx1250:
//   h_t = tanh(x_t*W_ih^T + b_ih + b_hh + h_{t-1} W_hh^T),  B=1024, T=1024, H=64
//
// Workgroup = 128 threads (4 wave32s) owning 16 batch rows for the whole
// recurrence. Wave w computes output columns N = [16w, 16w+16): 16
// v_wmma_f32_16x16x4_f32 per step as FOUR independent 4-deep K-chains.
// h ping-pongs through a padded LDS tile (C/D layout -> A layout transpose).
//
// Per-step sync is a SPLIT barrier: signal right after the LDS h_t stores
// (draining DScnt only), wait at the top of the next step after the
// independent x loads issued; NT output stores overlap the barrier round
// trip. All 16 A tiles are forced live at once via an asm register hold so
// the 8 ds_load_2addr_b64 issue as one clause with a single DScnt wait,
// followed by an uninterrupted 16-WMMA burst.

typedef __attribute__((ext_vector_type(2))) float v2f;
typedef __attribute__((ext_vector_type(8))) float v8f;

#define RNN_B 1024
#define RNN_T 1024
#define RNN_H 64
#define LDS_STRIDE 68   // 64 + 4 pad floats -> conflict-free LDS banks

__device__ __forceinline__ float fast_tanh(float x) {
#if __has_builtin(__builtin_amdgcn_tanhf)
    return __builtin_amdgcn_tanhf(x);           // v_tanh_f32
#else
    float r;
    asm volatile("v_tanh_f32 %0, %1\n\ts_nop 1" : "=v"(r) : "v"(x));
    return r;
#endif
}

// Split workgroup barrier, LDS-visibility only.
__device__ __forceinline__ void bar_signal_after_lds() {
    asm volatile("s_wait_dscnt 0x0\n\ts_barrier_signal -1" ::: "memory");
}
__device__ __forceinline__ void bar_wait() {
    asm volatile("s_barrier_wait -1" ::: "memory");
}

__global__ __launch_bounds__(128, 1) void rnn_tanh_wmma(
    const float* __restrict__ x,      // [B,T,1]
    const float* __restrict__ W_ih,   // [H,1]
    const float* __restrict__ b_ih,   // [H]
    const float* __restrict__ W_hh,   // [H,H]
    const float* __restrict__ b_hh,   // [H]
    float* __restrict__ out)          // [B,T,H]
{
    __shared__ __align__(16) float hbuf[2][16 * LDS_STRIDE];

    const int tid  = threadIdx.x;
    const int lane = tid & 31;        // lane within wave
    const int wv   = tid >> 5;        // wave id 0..3  -> N-tile
    const int j    = lane & 15;       // N index within tile / A row index
    const int half = lane >> 4;       // 0 or 1
    const int base = blockIdx.x * 16; // batch-row base for this workgroup
    const int ncol = wv * 16 + j;     // this lane's output column

    // ---- Preload this wave's W_hh B-operand tiles (B = W_hh^T, K x N) ----
    // Tile k covers K = 4k..4k+3, N = 16*wv..16*wv+15.
    //   lane j   : { W_hh[ncol][4k+0], W_hh[ncol][4k+1] }
    //   lane j+16: { W_hh[ncol][4k+2], W_hh[ncol][4k+3] }
    v2f Wt[16];
#pragma unroll
    for (int k = 0; k < 16; ++k) {
        const float* p = W_hh + ncol * RNN_H + (k * 4 + half * 2);
        Wt[k] = v2f{p[0], p[1]};
    }

    // Per-lane N is fixed in C/D layout -> one bias / input-weight scalar.
    const float bias = b_ih[ncol] + b_hh[ncol];
    const float wih  = W_ih[ncol];

    // h_{-1} = 0 in buffer 0; signal so the t=0 wait can pass.
    for (int i = tid; i < 16 * LDS_STRIDE; i += 128) hbuf[0][i] = 0.0f;
    bar_signal_after_lds();

    // C/D layout: VGPR r -> row M = half*8 + r.
    const float* xp = x + (size_t)(base + half * 8) * RNN_T;
    float* outp = out + (size_t)(base + half * 8) * RNN_T * RNN_H + ncol;

#pragma unroll 1
    for (int t = 0; t < RNN_T; ++t) {
        const float* cur = hbuf[t & 1];
        float*       nxt = hbuf[(t + 1) & 1];

        // x_t loads issue BEFORE the barrier wait (independent of h).
        float xv[8];
#pragma unroll
        for (int r = 0; r < 8; ++r) xv[r] = xp[(size_t)r * RNN_T + t];

        // Wait for all waves' h_{t-1} (signaled at the end of step t-1).
        bar_wait();

        // Load ALL A tiles (16x4 f32 slices of h_{t-1}).
        // A layout: lane j holds row M=j, K = 4k + half*2 .. +1 -> ds_load_b64
        v2f a[16];
#pragma unroll
        for (int k = 0; k < 16; ++k)
            a[k] = *(const v2f*)&cur[j * LDS_STRIDE + k * 4 + half * 2];

        // Force all 16 tiles live simultaneously: RA cannot recycle the load
        // destinations, and no WMMA can be hoisted above this point. The 8
        // ds_load_2addr_b64 issue as one clause with a single DScnt wait.
        asm volatile(""
            : "+v"(a[0]), "+v"(a[1]), "+v"(a[2]),  "+v"(a[3]),
              "+v"(a[4]), "+v"(a[5]), "+v"(a[6]),  "+v"(a[7]),
              "+v"(a[8]), "+v"(a[9]), "+v"(a[10]), "+v"(a[11]),
              "+v"(a[12]), "+v"(a[13]), "+v"(a[14]), "+v"(a[15]));

        // Four independent 4-deep accumulation chains (3 spare issue slots
        // between dependent WMMAs); fold input projection + biases into the
        // first chain's C operand.
        v8f acc0, acc1, acc2, acc3;
#pragma unroll
        for (int r = 0; r < 8; ++r) {
            acc0[r] = fmaf(xv[r], wih, bias);
            acc1[r] = 0.0f;
            acc2[r] = 0.0f;
            acc3[r] = 0.0f;
        }

#pragma unroll
        for (int k = 0; k < 4; ++k) {
            acc0 = __builtin_amdgcn_wmma_f32_16x16x4_f32(
                false, a[4 * k + 0], false, Wt[4 * k + 0], (short)0, acc0, false, false);
            acc1 = __builtin_amdgcn_wmma_f32_16x16x4_f32(
                false, a[4 * k + 1], false, Wt[4 * k + 1], (short)0, acc1, false, false);
            acc2 = __builtin_amdgcn_wmma_f32_16x16x4_f32(
                false, a[4 * k + 2], false, Wt[4 * k + 2], (short)0, acc2, false, false);
            acc3 = __builtin_amdgcn_wmma_f32_16x16x4_f32(
                false, a[4 * k + 3], false, Wt[4 * k + 3], (short)0, acc3, false, false);
        }

        // Pairwise combine (v_pk_add_f32), tanh, write h_t to LDS, signal.
        float hv[8];
#pragma unroll
        for (int r = 0; r < 8; ++r) {
            float s = (acc0[r] + acc1[r]) + (acc2[r] + acc3[r]);
            hv[r] = fast_tanh(s);
            nxt[(half * 8 + r) * LDS_STRIDE + ncol] = hv[r];
        }
        bar_signal_after_lds();

        // Output streaming overlaps the barrier round-trip + other waves.
#pragma unroll
        for (int r = 0; r < 8; ++r) {
            __builtin_nontemporal_store(
                hv[r], &outp[(size_t)r * RNN_T * RNN_H + (size_t)t * RNN_H]);
        }
    }

    // Consume the final signal so the barrier retires cleanly.
    bar_wait();
}

extern "C" void kernel_launch(void* const* d_in, const int* in_sizes, int n_in,
                              void* d_out, int out_size, void* d_ws, size_t ws_size,
                              hipStream_t stream) {
    const float* x    = (const float*)d_in[0];  // [B,T,1]
    const float* W_ih = (const float*)d_in[1];  // [H,1]
    const float* b_ih = (const float*)d_in[2];  // [H]
    const float* W_hh = (const float*)d_in[3];  // [H,H]
    const float* b_hh = (const float*)d_in[4];  // [H]
    float* out = (float*)d_out;                 // [B,T,H]

    rnn_tanh_wmma<<<RNN_B / 16, 128, 0, stream>>>(x, W_ih, b_ih, W_hh, b_hh, out);
}